// EdgeNet_14276471292017
// MI455X (gfx1250) — compile-verified
//
#include <hip/hip_runtime.h>
#include <math.h>

#define N_NODES 2048
#define N_EDGES 16384
#define D_FEAT  4

typedef __attribute__((ext_vector_type(2))) float v2f;
typedef __attribute__((ext_vector_type(8))) float v8f;

// ---------------------------------------------------------------------------
// Kernel 1: angles[e][0..3] = sum_n Ro[n][e]*X[n][d]   (bo)
//           angles[e][4..7] = sum_n Ri[n][e]*X[n][d]   (bi)
//
// Each wave32 computes a 32-edge tile pair (two 16x16x4 WMMA column tiles,
// contiguous 128B A-loads per row) over a 512-row K slice (split-K x4).
// The 4 K-slices of a tile pair live in one 256-thread block and are reduced
// through LDS. B operand (X) is staged in LDS pre-padded to 16 columns so
// all lanes load unconditionally (no exec predication in the hot loop).
// ---------------------------------------------------------------------------
__global__ __launch_bounds__(256) void edgenet_gemm_wmma(
    const float* __restrict__ X, const float* __restrict__ Ri,
    const float* __restrict__ Ro, float* __restrict__ angles)
{
  __shared__ float Xp[N_NODES * 16];          // 128 KB, cols 4..15 = 0
  __shared__ float red[2][4][32][8];          // 8 KB split-K partials

  for (int i = threadIdx.x; i < N_NODES * 16; i += blockDim.x) {
    const int r = i >> 4, cidx = i & 15;
    Xp[i] = (cidx < D_FEAT) ? X[r * D_FEAT + cidx] : 0.0f;
  }
  __syncthreads();

  const int wave   = threadIdx.x >> 5;
  const int lane   = threadIdx.x & 31;
  const int pl     = wave >> 2;               // pair-local: 0..1
  const int ksplit = wave & 3;                // split-K slice: 0..3
  const int pair   = blockIdx.x * 2 + pl;
  const int e0     = pair * 32;
  const int hi     = lane >> 4;               // half-wave K-group select
  const int col    = lane & 15;               // M for A, N for B/C/D
  const int rowoff = hi * 2;                  // lane covers K rows rowoff, rowoff+1
  const int kbase  = ksplit * (N_NODES / 4);  // 512-row slice

  const float* __restrict__ roP0 = Ro + (size_t)e0 + col;
  const float* __restrict__ roP1 = roP0 + 16;
  const float* __restrict__ riP0 = Ri + (size_t)e0 + col;
  const float* __restrict__ riP1 = riP0 + 16;

  v8f ao0 = {}, ao1 = {}, ai0 = {}, ai1 = {};

  size_t roff = (size_t)(kbase + rowoff) * N_EDGES;  // global row offset
  int    li   = (kbase + rowoff) * 16 + col;         // LDS B index

  #pragma unroll 4
  for (int n0 = 0; n0 < N_NODES / 4; n0 += 4) {
    v2f a, b;
    b.x = Xp[li];
    b.y = Xp[li + 16];
    // tile 0, Ro
    a.x = roP0[roff];
    a.y = roP0[roff + N_EDGES];
    ao0 = __builtin_amdgcn_wmma_f32_16x16x4_f32(false, a, false, b, (short)0, ao0, false, false);
    // tile 1, Ro
    a.x = roP1[roff];
    a.y = roP1[roff + N_EDGES];
    ao1 = __builtin_amdgcn_wmma_f32_16x16x4_f32(false, a, false, b, (short)0, ao1, false, false);
    // tile 0, Ri
    a.x = riP0[roff];
    a.y = riP0[roff + N_EDGES];
    ai0 = __builtin_amdgcn_wmma_f32_16x16x4_f32(false, a, false, b, (short)0, ai0, false, false);
    // tile 1, Ri
    a.x = riP1[roff];
    a.y = riP1[roff + N_EDGES];
    ai1 = __builtin_amdgcn_wmma_f32_16x16x4_f32(false, a, false, b, (short)0, ai1, false, false);

    roff += (size_t)4 * N_EDGES;
    li   += 4 * 16;
  }

  // Stash this wave's partials. D layout: VGPR r, lanes 0-15 -> D[r][lane],
  // lanes 16-31 -> D[r+8][lane-16].
  if (col < D_FEAT) {
    #pragma unroll
    for (int r = 0; r < 8; ++r) {
      const int m = hi * 8 + r;
      red[pl][ksplit][m][col]          = ao0[r];
      red[pl][ksplit][m][4 + col]      = ai0[r];
      red[pl][ksplit][16 + m][col]     = ao1[r];
      red[pl][ksplit][16 + m][4 + col] = ai1[r];
    }
  }
  __syncthreads();

  // Reduce the 4 split-K partials: 2 pairs * 32 edges * 8 angles = 512 values.
  for (int idx = threadIdx.x; idx < 512; idx += blockDim.x) {
    const int p = idx >> 8;          // pair-local
    const int e = (idx >> 3) & 31;   // edge within pair
    const int d = idx & 7;           // angle component
    const float v = red[p][0][e][d] + red[p][1][e][d] +
                    red[p][2][e][d] + red[p][3][e][d];
    angles[(size_t)((blockIdx.x * 2 + p) * 32 + e) * 8 + d] = v;
  }
}

// ---------------------------------------------------------------------------
// Kernel 2: 8-qubit statevector circuit, one wave32 per edge.
// Qubits 0..4 -> lane bits (q0 = lane bit4 ... q4 = lane bit0).
// Qubits 5..7 -> local state index bits (q5 = bit2, q6 = bit1, q7 = bit0).
// Trig uses the native v_sin_f32/v_cos_f32 transcendental units.
// ---------------------------------------------------------------------------
template<int SH>   // RY on a lane-bit wire; SH = 4 - wire
__device__ __forceinline__ void ry_lane(float (&st)[8], int lane, float gc, float gs) {
  const float sgn = ((lane >> SH) & 1) ? gs : -gs;
  #pragma unroll
  for (int i = 0; i < 8; ++i) {
    float o = __shfl_xor(st[i], 1 << SH, 32);
    st[i] = gc * st[i] + sgn * o;
  }
}

template<int M>    // RY on a local wire; M = 1 << (7 - wire)
__device__ __forceinline__ void ry_local(float (&st)[8], float gc, float gs) {
  #pragma unroll
  for (int i = 0; i < 8; ++i)
    if (!(i & M)) {
      float a0 = st[i], a1 = st[i + M];
      st[i]     = gc * a0 - gs * a1;
      st[i + M] = gs * a0 + gc * a1;
    }
}

template<int CSH, int TSH>  // CX: control + target both lane bits
__device__ __forceinline__ void cx_ll(float (&st)[8], int lane) {
  const bool cb = (lane >> CSH) & 1;
  #pragma unroll
  for (int i = 0; i < 8; ++i) {
    float o = __shfl_xor(st[i], 1 << TSH, 32);
    st[i] = cb ? o : st[i];
  }
}

template<int CSH, int TM>   // CX: control lane bit, target local bit
__device__ __forceinline__ void cx_lane_local(float (&st)[8], int lane) {
  const bool cb = (lane >> CSH) & 1;
  #pragma unroll
  for (int i = 0; i < 8; ++i)
    if (!(i & TM)) {
      float a = st[i], b = st[i + TM];
      st[i]      = cb ? b : a;
      st[i + TM] = cb ? a : b;
    }
}

template<int CM, int TM>    // CX: control + target both local bits
__device__ __forceinline__ void cx_local(float (&st)[8]) {
  #pragma unroll
  for (int i = 0; i < 8; ++i)
    if ((i & CM) && !(i & TM)) {
      float t = st[i]; st[i] = st[i + TM]; st[i + TM] = t;
    }
}

__global__ __launch_bounds__(256) void edgenet_circuit(
    const float* __restrict__ angles, const float* __restrict__ theta,
    float* __restrict__ out)
{
  const int wave = threadIdx.x >> 5;
  const int lane = threadIdx.x & 31;
  const int edge = blockIdx.x * 8 + wave;

  // Per-qubit initial amplitudes from the edge's 8 angles.
  float c[8], s[8];
  #pragma unroll
  for (int q = 0; q < 8; ++q) {
    float a = 0.5f * angles[(size_t)edge * 8 + q];
    c[q] = __cosf(a);
    s[q] = __sinf(a);
  }

  // Product state: lane bits pick q0..q4, local index picks q5..q7.
  const int b0 = (lane >> 4) & 1, b1 = (lane >> 3) & 1, b2 = (lane >> 2) & 1;
  const int b3 = (lane >> 1) & 1, b4 = lane & 1;
  const float laneAmp = (b0 ? s[0] : c[0]) * (b1 ? s[1] : c[1]) *
                        (b2 ? s[2] : c[2]) * (b3 ? s[3] : c[3]) *
                        (b4 ? s[4] : c[4]);
  float st[8];
  #pragma unroll
  for (int i = 0; i < 8; ++i) {
    const int q5 = (i >> 2) & 1, q6 = (i >> 1) & 1, q7 = i & 1;
    st[i] = laneAmp * (q5 ? s[5] : c[5]) * (q6 ? s[6] : c[6]) * (q7 ? s[7] : c[7]);
  }

  float tc[21], ts[21];
  #pragma unroll
  for (int i = 0; i < 21; ++i) {
    float a = 0.5f * theta[i];
    tc[i] = __cosf(a);
    ts[i] = __sinf(a);
  }

  // OPS sequence (wire->template: lane SH = 4-w; local M = 1<<(7-w))
  ry_lane<3>(st, lane, tc[0],  ts[0]);   // ry w1
  ry_lane<2>(st, lane, tc[1],  ts[1]);   // ry w2
  cx_ll<3, 2>(st, lane);                 // cx 1,2
  ry_local<4>(st, tc[2],  ts[2]);        // ry w5
  ry_local<2>(st, tc[3],  ts[3]);        // ry w6
  cx_local<2, 4>(st);                    // cx 6,5
  ry_lane<4>(st, lane, tc[4],  ts[4]);   // ry w0
  ry_lane<3>(st, lane, tc[5],  ts[5]);   // ry w1
  cx_ll<4, 3>(st, lane);                 // cx 0,1
  ry_lane<2>(st, lane, tc[6],  ts[6]);   // ry w2
  ry_lane<1>(st, lane, tc[7],  ts[7]);   // ry w3
  cx_ll<1, 2>(st, lane);                 // cx 3,2
  ry_lane<0>(st, lane, tc[8],  ts[8]);   // ry w4
  ry_local<4>(st, tc[9],  ts[9]);        // ry w5
  cx_lane_local<0, 4>(st, lane);         // cx 4,5
  ry_local<2>(st, tc[10], ts[10]);       // ry w6
  ry_local<1>(st, tc[11], ts[11]);       // ry w7
  cx_local<1, 2>(st);                    // cx 7,6
  ry_lane<2>(st, lane, tc[12], ts[12]);  // ry w2
  ry_local<4>(st, tc[13], ts[13]);       // ry w5
  cx_lane_local<2, 4>(st, lane);         // cx 2,5
  ry_lane<3>(st, lane, tc[14], ts[14]);  // ry w1
  ry_lane<2>(st, lane, tc[15], ts[15]);  // ry w2
  cx_ll<3, 2>(st, lane);                 // cx 1,2
  ry_local<4>(st, tc[16], ts[16]);       // ry w5
  ry_local<2>(st, tc[17], ts[17]);       // ry w6
  cx_local<2, 4>(st);                    // cx 6,5
  ry_lane<2>(st, lane, tc[18], ts[18]);  // ry w2
  ry_local<4>(st, tc[19], ts[19]);       // ry w5
  cx_lane_local<2, 4>(st, lane);         // cx 2,5
  ry_local<4>(st, tc[20], ts[20]);       // ry w5

  // <Z> on wire 5 (local bit2): diff = P(q5=0) - P(q5=1), out = (1-diff)/2
  float diff = 0.0f;
  #pragma unroll
  for (int i = 0; i < 4; ++i) diff += st[i] * st[i];
  #pragma unroll
  for (int i = 4; i < 8; ++i) diff -= st[i] * st[i];
  #pragma unroll
  for (int off = 16; off >= 1; off >>= 1)
    diff += __shfl_xor(diff, off, 32);

  if (lane == 0)
    out[edge] = (1.0f - diff) * 0.5f;
}

// ---------------------------------------------------------------------------
extern "C" void kernel_launch(void* const* d_in, const int* in_sizes, int n_in,
                              void* d_out, int out_size, void* d_ws, size_t ws_size,
                              hipStream_t stream) {
  const float* X     = (const float*)d_in[0];
  const float* Ri    = (const float*)d_in[1];
  const float* Ro    = (const float*)d_in[2];
  const float* theta = (const float*)d_in[3];
  float* angles = (float*)d_ws;            // 16384 * 8 floats = 512 KB scratch
  float* out    = (float*)d_out;

  // 512 tile pairs (32 edges each) * 4 split-K waves -> 2048 waves, 256 blocks
  edgenet_gemm_wmma<<<256, 256, 0, stream>>>(X, Ri, Ro, angles);
  // one wave per edge, 8 waves per block -> 2048 blocks
  edgenet_circuit<<<N_EDGES / 8, 256, 0, stream>>>(angles, theta, out);
}